// DigitCaps_53687091200485
// MI455X (gfx1250) — compile-verified
//
#include <hip/hip_runtime.h>
#include <hip/hip_bf16.h>
#include <math.h>

#define BATCH 128
#define CIN   1152
#define IDIM  8
#define OCAPS 10
#define ODIM  16
#define OD    160   // OCAPS*ODIM

typedef __attribute__((ext_vector_type(2))) float v2f;
typedef __attribute__((ext_vector_type(8))) float v8f;

// ---------------------------------------------------------------------------
// Kernel 1: u_hat[b][c][o][d] = sum_i W[o][c][d][i] * x[b][c][i]
// One wave per (batch-tile of 16, o, strip of 16 c). Per c: two chained
// V_WMMA_F32_16X16X4_F32 (K=0..3, K=4..7). M=batch, N=OUT_DIM, K=IN_DIM.
// ---------------------------------------------------------------------------
__global__ __launch_bounds__(256) void caps_uhat_wmma(
    const float* __restrict__ x, const float* __restrict__ W,
    float* __restrict__ uhat)
{
  const int lane  = threadIdx.x & 31;
  const int wave  = (blockIdx.x * 256 + threadIdx.x) >> 5;  // 0..5759
  const int cg    = wave % 72;          // 72 strips of 16 c
  const int o     = (wave / 72) % OCAPS;
  const int bt    = wave / 720;         // 0..7 batch tiles
  const int c0    = cg * 16;
  const int n     = lane & 15;          // A: M row = lane%16; B/D: N col = lane%16
  const int khalf = (lane >> 4) << 1;   // 0 (lanes 0-15) or 2 (lanes 16-31)
  const int mh    = (lane >> 4) << 3;   // D-matrix M offset: 0 or 8

  for (int cc = 0; cc < 16; ++cc) {
    const int c = c0 + cc;
    // A tile: x[bt*16 + m, c, k]  (m = lane%16, k = khalf / khalf+1, +4 for step 2)
    const size_t xoff = ((size_t)(bt * 16 + n) * CIN + c) * IDIM + khalf;
    v2f a0 = *(const v2f*)(x + xoff);
    v2f a1 = *(const v2f*)(x + xoff + 4);
    // B tile: B[k][n] = W[o, c, d=n, i=k]
    const size_t woff = (((size_t)o * CIN + c) * ODIM + n) * IDIM + khalf;
    v2f b0 = *(const v2f*)(W + woff);
    v2f b1 = *(const v2f*)(W + woff + 4);

    v8f acc = {};
    acc = __builtin_amdgcn_wmma_f32_16x16x4_f32(false, a0, false, b0,
                                                (short)0, acc, false, false);
    acc = __builtin_amdgcn_wmma_f32_16x16x4_f32(false, a1, false, b1,
                                                (short)0, acc, false, false);
    // D layout: VGPR j holds M = mh + j, N = lane%16
    #pragma unroll
    for (int j = 0; j < 8; ++j) {
      const int brow = bt * 16 + mh + j;
      uhat[((size_t)brow * CIN + c) * OD + o * ODIM + n] = acc[j];
    }
  }
}

// ---------------------------------------------------------------------------
// Kernel 2: fused 3-iteration dynamic routing. One block per batch element.
// b-logits in LDS; per-c softmax stats cached; phase-2 u_hat reads coalesced.
// ---------------------------------------------------------------------------
__global__ __launch_bounds__(320) void caps_routing(
    const float* __restrict__ uhat, float* __restrict__ out)
{
  __shared__ float Blds[OCAPS * CIN];   // routing logits b[o][c]      (45 KB)
  __shared__ float Mlds[CIN];           // per-c softmax max over o
  __shared__ float Sinv[CIN];           // per-c 1/sum(exp)
  __shared__ float partial[320];
  __shared__ float slds[OD];
  __shared__ float vlds[OD];

  const int tid = threadIdx.x;
  const int b   = blockIdx.x;

  for (int i = tid; i < OCAPS * CIN; i += 320) Blds[i] = 0.f;
  for (int i = tid; i < CIN; i += 320) { Mlds[i] = 0.f; Sinv[i] = 0.1f; }
  __syncthreads();   // stats now encode softmax(0) = uniform 0.1 for t==0

  for (int t = 0; t < 3; ++t) {
    if (t > 0) {
      // Phase 1 (lane-per-c): b[o,c] += <u_hat[b,c,o,:], v[o,:]>, then
      // softmax stats over the 10 o values (all lane-local).
      #pragma unroll
      for (int k = 0; k < 4; ++k) {
        const int c = tid + k * 320;
        if (c < CIN) {
          const float4* up4 = (const float4*)(uhat + ((size_t)b * CIN + c) * OD);
          float nb[OCAPS];
          #pragma unroll
          for (int o = 0; o < OCAPS; ++o) {
            float acc = 0.f;
            #pragma unroll
            for (int q = 0; q < 4; ++q) {
              const float4 u = up4[o * 4 + q];
              const int vb = o * 16 + q * 4;
              acc += u.x * vlds[vb] + u.y * vlds[vb + 1]
                   + u.z * vlds[vb + 2] + u.w * vlds[vb + 3];
            }
            const float nbv = Blds[o * CIN + c] + acc;
            Blds[o * CIN + c] = nbv;
            nb[o] = nbv;
          }
          float mx = nb[0];
          #pragma unroll
          for (int o = 1; o < OCAPS; ++o) mx = fmaxf(mx, nb[o]);
          float se = 0.f;
          #pragma unroll
          for (int o = 0; o < OCAPS; ++o) se += __expf(nb[o] - mx);
          Mlds[c] = mx;
          Sinv[c] = 1.0f / se;
        }
      }
    }
    __syncthreads();

    // Phase 2: s[o,d] = sum_c softmax(b)[o,c] * u_hat[b,c,o,d]
    // threads = (c-half g in {0,1}) x (o*16+d); global reads fully coalesced.
    {
      const int g    = tid / 160;
      const int idx  = tid - g * 160;   // o*16 + d
      const int o    = idx >> 4;
      const int cbeg = g * 576;
      const float* up = uhat + ((size_t)b * CIN + cbeg) * OD + idx;
      float sacc = 0.f;
      #pragma unroll 4
      for (int j = 0; j < 576; ++j) {
        const int c = cbeg + j;
        const float ccv = __expf(Blds[o * CIN + c] - Mlds[c]) * Sinv[c];
        sacc += ccv * up[(size_t)j * OD];
      }
      partial[tid] = sacc;
    }
    __syncthreads();
    if (tid < OD) slds[tid] = partial[tid] + partial[tid + 160];
    __syncthreads();
    if (tid < OCAPS) {  // squash: (||s||^2/(1+||s||^2)) * s / sqrt(||s||^2+eps)
      float sq = 0.f;
      #pragma unroll
      for (int d = 0; d < ODIM; ++d) { const float sv = slds[tid * 16 + d]; sq += sv * sv; }
      const float scale = (sq / (1.f + sq)) * rsqrtf(sq + 1e-8f);
      #pragma unroll
      for (int d = 0; d < ODIM; ++d) vlds[tid * 16 + d] = scale * slds[tid * 16 + d];
    }
    __syncthreads();
  }

  if (tid < OD) out[(size_t)b * OD + tid] = vlds[tid];
}

// ---------------------------------------------------------------------------
extern "C" void kernel_launch(void* const* d_in, const int* in_sizes, int n_in,
                              void* d_out, int out_size, void* d_ws, size_t ws_size,
                              hipStream_t stream) {
  (void)in_sizes; (void)n_in; (void)out_size; (void)ws_size;
  const float* x = (const float*)d_in[0];   // [128,1152,8]
  const float* W = (const float*)d_in[1];   // [10,1152,16,8]
  float* out  = (float*)d_out;              // [128,10,16]
  float* uhat = (float*)d_ws;               // [128,1152,10,16] f32, ~90 MB (L2-resident)

  // 8 batch-tiles * 10 o * 72 c-strips = 5760 waves = 720 blocks of 8 waves
  caps_uhat_wmma<<<720, 256, 0, stream>>>(x, W, uhat);
  caps_routing<<<BATCH, 320, 0, stream>>>(uhat, out);
}